// StructureTransformer_23476291240035
// MI455X (gfx1250) — compile-verified
//
#include <hip/hip_runtime.h>

// ---------------------------------------------------------------------------
// StructureTransformer for MI455X (gfx1250, wave32).
// All heavy GEMMs use V_WMMA_F32_16X16X4_F32 (full fp32 fidelity).
// Workspace layout (bytes):
//   [0)              E_idx  : int  [8*1024*30]            =    983,040
//   [983040)         hv0    : f32  [8*1024*128]           =  4,194,304
//   [5177344)        h_E    : f32  [8*1024*30*128]        = 125,829,120
//   total ~ 131 MB
// ---------------------------------------------------------------------------

typedef __attribute__((ext_vector_type(2))) float v2f;
typedef __attribute__((ext_vector_type(8))) float v8f;

#define NB  8
#define LSEQ 1024
#define KTOP 30
#define HID 128
#define EINF 416

__device__ inline v8f wmma4(v2f a, v2f b, v8f c) {
  return __builtin_amdgcn_wmma_f32_16x16x4_f32(false, a, false, b, (short)0, c, false, false);
}

// ---------------- small vec helpers ----------------
struct F3 { float x, y, z; };
__device__ inline F3 f3sub(F3 a, F3 b) { return {a.x-b.x, a.y-b.y, a.z-b.z}; }
__device__ inline F3 f3cross(F3 a, F3 b) {
  return {a.y*b.z - a.z*b.y, a.z*b.x - a.x*b.z, a.x*b.y - a.y*b.x};
}
__device__ inline float f3dot(F3 a, F3 b) { return a.x*b.x + a.y*b.y + a.z*b.z; }
__device__ inline F3 f3norm(F3 a) {
  float inv = 1.0f / (sqrtf(f3dot(a, a)) + 1e-8f);
  return {a.x*inv, a.y*inv, a.z*inv};
}

// ---------------------------------------------------------------------------
// Kernel 1: per (b,i) compute masked distances to all j, replicate
// Dadj = D + (1-mask2)*Dmax, then 30x argmin with lower-index tie-break
// (matches jax.lax.top_k on -Dadj).
// ---------------------------------------------------------------------------
__global__ __launch_bounds__(128) void k_topk(const float* __restrict__ X,
                                              const float* __restrict__ mask,
                                              int* __restrict__ E_idx) {
  __shared__ float d[LSEQ];
  __shared__ float redv[128];
  __shared__ int   redi[128];
  __shared__ float s_dmax;
  int bi = blockIdx.x;
  int b = bi >> 10, i = bi & (LSEQ - 1);
  int tid = threadIdx.x;
  const float* Xb = X + (size_t)b * LSEQ * 12;
  float cix = Xb[i*12 + 3], ciy = Xb[i*12 + 4], ciz = Xb[i*12 + 5]; // Ca = atom 1
  float mi = mask[b*LSEQ + i];
  float lmax = -1e30f;
  for (int j = tid; j < LSEQ; j += 128) {
    float dx = cix - Xb[j*12 + 3];
    float dy = ciy - Xb[j*12 + 4];
    float dz = ciz - Xb[j*12 + 5];
    float m2 = mi * mask[b*LSEQ + j];
    float dd = m2 * sqrtf(dx*dx + dy*dy + dz*dz + 1e-6f);
    d[j] = dd;
    lmax = fmaxf(lmax, dd);
  }
  redv[tid] = lmax; __syncthreads();
  for (int s = 64; s > 0; s >>= 1) { if (tid < s) redv[tid] = fmaxf(redv[tid], redv[tid+s]); __syncthreads(); }
  if (tid == 0) s_dmax = redv[0];
  __syncthreads();
  float dmax = s_dmax;
  for (int j = tid; j < LSEQ; j += 128) {
    float m2 = mi * mask[b*LSEQ + j];
    d[j] += (1.0f - m2) * dmax;
  }
  __syncthreads();
  for (int k = 0; k < KTOP; ++k) {
    float bv = 1e30f; int bx = LSEQ;
    for (int j = tid; j < LSEQ; j += 128) {
      float v = d[j];
      if (v < bv) { bv = v; bx = j; }
    }
    redv[tid] = bv; redi[tid] = bx; __syncthreads();
    for (int s = 64; s > 0; s >>= 1) {
      if (tid < s) {
        float ov = redv[tid+s]; int oi = redi[tid+s];
        if (ov < redv[tid] || (ov == redv[tid] && oi < redi[tid])) { redv[tid] = ov; redi[tid] = oi; }
      }
      __syncthreads();
    }
    if (tid == 0) {
      E_idx[(size_t)bi*KTOP + k] = redi[0];
      d[redi[0]] = 1e30f;
    }
    __syncthreads();
  }
}

// ---------------------------------------------------------------------------
// Kernel 2: dihedral node features (6) -> node_W/LN -> @Wv -> hv0 (B,L,128)
// ---------------------------------------------------------------------------
__device__ inline F3 xb_load(const float* X, int b, int v) {
  const float* p = X + ((size_t)(b*LSEQ + (v/3))*4 + (v%3))*3;
  return {p[0], p[1], p[2]};
}

__global__ __launch_bounds__(128) void k_node(const float* __restrict__ X,
    const float* __restrict__ node_W, const float* __restrict__ node_b,
    const float* __restrict__ node_g, const float* __restrict__ node_beta,
    const float* __restrict__ Wv, const float* __restrict__ Wv_b,
    float* __restrict__ hv0) {
  __shared__ float feat[6];
  __shared__ float vln[HID];
  __shared__ float red[128];
  int bi = blockIdx.x;
  int b = bi >> 10, i = bi & (LSEQ - 1);
  int tid = threadIdx.x;
  if (tid < 3) {
    int m = 3*i + tid;
    float Dang = 0.0f;
    const int total = 3 * LSEQ;
    if (m >= 1 && m <= total - 3) {
      int u = m - 1;
      F3 p0 = xb_load(X, b, u),   p1 = xb_load(X, b, u+1);
      F3 p2 = xb_load(X, b, u+2), p3 = xb_load(X, b, u+3);
      F3 u2 = f3norm(f3sub(p1, p0));
      F3 u1 = f3norm(f3sub(p2, p1));
      F3 u0 = f3norm(f3sub(p3, p2));
      F3 n2 = f3norm(f3cross(u2, u1));
      F3 n1 = f3norm(f3cross(u1, u0));
      float cosD = fminf(fmaxf(f3dot(n2, n1), -1.0f + 1e-7f), 1.0f - 1e-7f);
      float s = f3dot(u2, n1);
      float sg = (s > 0.0f) ? 1.0f : ((s < 0.0f) ? -1.0f : 0.0f);
      Dang = sg * acosf(cosD);
    }
    feat[tid]   = cosf(Dang);
    feat[3+tid] = sinf(Dang);
  }
  __syncthreads();
  float acc = node_b[tid];
  #pragma unroll
  for (int c = 0; c < 6; ++c) acc += feat[c] * node_W[c*HID + tid];
  red[tid] = acc; __syncthreads();
  for (int s = 64; s > 0; s >>= 1) { if (tid < s) red[tid] += red[tid+s]; __syncthreads(); }
  float mu = red[0] * (1.0f/HID); __syncthreads();
  float dx = acc - mu;
  red[tid] = dx*dx; __syncthreads();
  for (int s = 64; s > 0; s >>= 1) { if (tid < s) red[tid] += red[tid+s]; __syncthreads(); }
  float var = red[0] * (1.0f/HID); __syncthreads();
  vln[tid] = dx * rsqrtf(var + 1e-5f) * node_g[tid] + node_beta[tid];
  __syncthreads();
  float o = Wv_b[tid];
  for (int c = 0; c < HID; ++c) o += vln[c] * Wv[c*HID + tid];
  hv0[(size_t)bi*HID + tid] = o;
}

// ---------------------------------------------------------------------------
// Kernel 3 (fused, WMMA): per (b,i): build 30x416 edge features in LDS,
// GEMM (32x416 @ 416x128) + bias -> LN -> GEMM (32x128 @ 128x128) + bias
// -> h_E (B,L,K,128).
// ---------------------------------------------------------------------------
__device__ inline void calc_atoms(const float* Xr, float* out /*15: Ca,N,C,O,Cb*/) {
  float Nx = Xr[0], Ny = Xr[1], Nz = Xr[2];
  float Ax = Xr[3], Ay = Xr[4], Az = Xr[5];  // Ca
  float Cx = Xr[6], Cy = Xr[7], Cz = Xr[8];
  float Ox = Xr[9], Oy = Xr[10], Oz = Xr[11];
  float bx = Ax - Nx, by = Ay - Ny, bz = Az - Nz;
  float cx = Cx - Ax, cy = Cy - Ay, cz = Cz - Az;
  float ax = by*cz - bz*cy, ay = bz*cx - bx*cz, az = bx*cy - by*cx;
  float Bx = -0.58273431f*ax + 0.56802827f*bx - 0.54067466f*cx + Ax;
  float By = -0.58273431f*ay + 0.56802827f*by - 0.54067466f*cy + Ay;
  float Bz = -0.58273431f*az + 0.56802827f*bz - 0.54067466f*cz + Az;
  out[0]=Ax; out[1]=Ay; out[2]=Az;
  out[3]=Nx; out[4]=Ny; out[5]=Nz;
  out[6]=Cx; out[7]=Cy; out[8]=Cz;
  out[9]=Ox; out[10]=Oy; out[11]=Oz;
  out[12]=Bx; out[13]=By; out[14]=Bz;
}

__global__ __launch_bounds__(256) void k_edge(const float* __restrict__ X,
    const int* __restrict__ E_idx, const int* __restrict__ srr,
    const float* __restrict__ pos_W, const float* __restrict__ pos_b,
    const float* __restrict__ edge_W, const float* __restrict__ edge_b,
    const float* __restrict__ edge_g, const float* __restrict__ edge_beta,
    const float* __restrict__ We, const float* __restrict__ We_b,
    float* __restrict__ h_E) {
  __shared__ float buf[32*EINF];   // 53,248 B; re-used (aliased) as E tile 32x128
  __shared__ float nba[KTOP*15];
  __shared__ float ctr[15];
  __shared__ float dall[KTOP*25];
  int bi = blockIdx.x;
  int b = bi >> 10, i = bi & (LSEQ - 1);
  int tid = threadIdx.x;

  if (tid < KTOP) {
    int j = E_idx[(size_t)bi*KTOP + tid];
    calc_atoms(X + (size_t)(b*LSEQ + j)*12, &nba[tid*15]);
  } else if (tid == KTOP) {
    calc_atoms(X + (size_t)(b*LSEQ + i)*12, ctr);
  }
  __syncthreads();
  for (int t = tid; t < KTOP*25; t += 256) {
    int k = t / 25, pr = t % 25, a = pr / 5, p = pr % 5;
    float dx = ctr[a*3+0] - nba[k*15 + p*3+0];
    float dy = ctr[a*3+1] - nba[k*15 + p*3+1];
    float dz = ctr[a*3+2] - nba[k*15 + p*3+2];
    dall[t] = sqrtf(dx*dx + dy*dy + dz*dz + 1e-6f);
  }
  __syncthreads();
  int ri = srr[b*LSEQ + i];
  for (int t = tid; t < 32*EINF; t += 256) {
    int k = t / EINF, c = t % EINF;
    float val = 0.0f;
    if (k < KTOP) {
      if (c < 16) {
        int j = E_idx[(size_t)bi*KTOP + k];
        int off = ri - srr[b*LSEQ + j];
        off = min(max(off, -32), 32) + 32;
        val = pos_W[off*16 + c] + pos_b[c];
      } else {
        int cc = c - 16, pr = cc >> 4, r = cc & 15;
        float mu = 2.0f + (20.0f/15.0f) * (float)r;
        float z = (dall[k*25 + pr] - mu) * 0.8f;   // sigma = 1.25
        val = __expf(-z*z);
      }
    }
    buf[t] = val;
  }
  __syncthreads();

  int lane = tid & 31, wv = tid >> 5;
  int hf = lane >> 4, lm = lane & 15;
  int n0 = wv * 16;

  // GEMM1: (32 x 416) @ edge_W (416 x 128)
  v8f c0 = {0,0,0,0,0,0,0,0}, c1 = {0,0,0,0,0,0,0,0};
  for (int k0 = 0; k0 < EINF; k0 += 4) {
    int kk = k0 + 2*hf;
    v2f bf; bf.x = edge_W[(size_t)kk*HID + n0 + lm];
            bf.y = edge_W[(size_t)(kk+1)*HID + n0 + lm];
    v2f a0; a0.x = buf[lm*EINF + kk];       a0.y = buf[lm*EINF + kk + 1];
    v2f a1; a1.x = buf[(16+lm)*EINF + kk];  a1.y = buf[(16+lm)*EINF + kk + 1];
    c0 = wmma4(a0, bf, c0);
    c1 = wmma4(a1, bf, c1);
  }
  __syncthreads();
  float eb = edge_b[n0 + lm];
  #pragma unroll
  for (int r = 0; r < 8; ++r) {
    buf[(r + 8*hf)*HID + n0 + lm]        = c0[r] + eb;
    buf[(16 + r + 8*hf)*HID + n0 + lm]   = c1[r] + eb;
  }
  __syncthreads();
  // LayerNorm rows 0..29
  if (tid < KTOP) {
    float mu = 0.0f;
    for (int c = 0; c < HID; ++c) mu += buf[tid*HID + c];
    mu *= (1.0f/HID);
    float var = 0.0f;
    for (int c = 0; c < HID; ++c) { float dv = buf[tid*HID + c] - mu; var += dv*dv; }
    var *= (1.0f/HID);
    float inv = rsqrtf(var + 1e-5f);
    for (int c = 0; c < HID; ++c)
      buf[tid*HID + c] = (buf[tid*HID + c] - mu) * inv * edge_g[c] + edge_beta[c];
  }
  __syncthreads();
  // GEMM2: (32 x 128) @ We (128 x 128)
  v8f d0 = {0,0,0,0,0,0,0,0}, d1 = {0,0,0,0,0,0,0,0};
  for (int k0 = 0; k0 < HID; k0 += 4) {
    int kk = k0 + 2*hf;
    v2f bf; bf.x = We[(size_t)kk*HID + n0 + lm];
            bf.y = We[(size_t)(kk+1)*HID + n0 + lm];
    v2f a0; a0.x = buf[lm*HID + kk];       a0.y = buf[lm*HID + kk + 1];
    v2f a1; a1.x = buf[(16+lm)*HID + kk];  a1.y = buf[(16+lm)*HID + kk + 1];
    d0 = wmma4(a0, bf, d0);
    d1 = wmma4(a1, bf, d1);
  }
  float wb = We_b[n0 + lm];
  #pragma unroll
  for (int r = 0; r < 8; ++r) {
    int row0 = r + 8*hf;
    if (row0 < KTOP) h_E[((size_t)bi*KTOP + row0)*HID + n0 + lm] = d0[r] + wb;
    int row1 = 16 + r + 8*hf;
    if (row1 < KTOP) h_E[((size_t)bi*KTOP + row1)*HID + n0 + lm] = d1[r] + wb;
  }
}

// ---------------------------------------------------------------------------
// Kernel 4 (fused layer, WMMA): per (b,i): gather h_EV(32x256) in LDS,
// Q = hV@WQ, K = h_EV@WK (WMMA), softmax(QK/sqrt d), V = h_EV@WVp (WMMA),
// dh = attn@V @ WO, LN residual, FFN 128->512->128, LN residual -> hv_out.
// ---------------------------------------------------------------------------
__device__ inline void block_ln_256(float* arr, const float* g, const float* bta,
                                    float* red, int tid) {
  float x = (tid < HID) ? arr[tid] : 0.0f;
  red[tid] = x; __syncthreads();
  for (int s = 128; s > 0; s >>= 1) { if (tid < s) red[tid] += red[tid+s]; __syncthreads(); }
  float mu = red[0] * (1.0f/HID); __syncthreads();
  float dx = (tid < HID) ? (x - mu) : 0.0f;
  red[tid] = dx*dx; __syncthreads();
  for (int s = 128; s > 0; s >>= 1) { if (tid < s) red[tid] += red[tid+s]; __syncthreads(); }
  float var = red[0] * (1.0f/HID); __syncthreads();
  if (tid < HID) arr[tid] = dx * rsqrtf(var + 1e-5f) * g[tid] + bta[tid];
  __syncthreads();
}

__global__ __launch_bounds__(256) void k_layer(const float* __restrict__ hv_in,
    const float* __restrict__ h_E, const int* __restrict__ E_idx,
    const float* __restrict__ mask,
    const float* __restrict__ WQ, const float* __restrict__ WK,
    const float* __restrict__ WVp, const float* __restrict__ WO,
    const float* __restrict__ n1g, const float* __restrict__ n1b,
    const float* __restrict__ Win, const float* __restrict__ Win_b,
    const float* __restrict__ Wout, const float* __restrict__ Wout_b,
    const float* __restrict__ n2g, const float* __restrict__ n2b,
    float* __restrict__ hv_out) {
  __shared__ float hEV[32*256];   // 32 KB
  __shared__ float kv[32*HID];    // 16 KB (K, then reused for V)
  __shared__ float Q[HID];
  __shared__ float att[4][32];
  __shared__ float dh[HID];
  __shared__ float hmid[HID];
  __shared__ float uarr[4*HID];
  __shared__ float red[256];
  int bi = blockIdx.x;
  int b = bi >> 10, i = bi & (LSEQ - 1);
  int tid = threadIdx.x;

  for (int t = tid; t < 32*256; t += 256) {
    int k = t >> 8, c = t & 255;
    float v = 0.0f;
    if (k < KTOP) {
      if (c < HID) v = h_E[((size_t)bi*KTOP + k)*HID + c];
      else {
        int j = E_idx[(size_t)bi*KTOP + k];
        v = hv_in[((size_t)(b*LSEQ + j))*HID + (c - HID)];
      }
    }
    hEV[t] = v;
  }
  __syncthreads();
  if (tid < HID) {
    const float* hvp = hv_in + (size_t)bi*HID;
    float acc = 0.0f;
    for (int c = 0; c < HID; ++c) acc += hvp[c] * WQ[c*HID + tid];
    Q[tid] = acc;
  }

  int lane = tid & 31, wv = tid >> 5;
  int hf = lane >> 4, lm = lane & 15;
  int n0 = wv * 16;

  // K = hEV (32x256) @ WK (256x128)
  v8f c0 = {0,0,0,0,0,0,0,0}, c1 = {0,0,0,0,0,0,0,0};
  for (int k0 = 0; k0 < 256; k0 += 4) {
    int kk = k0 + 2*hf;
    v2f bf; bf.x = WK[(size_t)kk*HID + n0 + lm];
            bf.y = WK[(size_t)(kk+1)*HID + n0 + lm];
    v2f a0; a0.x = hEV[lm*256 + kk];      a0.y = hEV[lm*256 + kk + 1];
    v2f a1; a1.x = hEV[(16+lm)*256 + kk]; a1.y = hEV[(16+lm)*256 + kk + 1];
    c0 = wmma4(a0, bf, c0);
    c1 = wmma4(a1, bf, c1);
  }
  __syncthreads();
  #pragma unroll
  for (int r = 0; r < 8; ++r) {
    kv[(r + 8*hf)*HID + n0 + lm]      = c0[r];
    kv[(16 + r + 8*hf)*HID + n0 + lm] = c1[r];
  }
  __syncthreads();
  // logits
  if (tid < 4*KTOP) {
    int h = tid / KTOP, k = tid % KTOP;
    float acc = 0.0f;
    for (int d = 0; d < 32; ++d) acc += Q[h*32 + d] * kv[k*HID + h*32 + d];
    att[h][k] = acc * 0.17677669529663687f;   // 1/sqrt(32)
  }
  __syncthreads();
  // softmax with mask
  if (tid < 4) {
    int h = tid;
    float mi = mask[b*LSEQ + i];
    float lg[KTOP], mk[KTOP];
    float mx = -1e30f;
    for (int k = 0; k < KTOP; ++k) {
      int j = E_idx[(size_t)bi*KTOP + k];
      mk[k] = mi * mask[b*LSEQ + j];
      lg[k] = (mk[k] > 0.0f) ? att[h][k] : -1e9f;
      mx = fmaxf(mx, lg[k]);
    }
    float s = 0.0f;
    for (int k = 0; k < KTOP; ++k) { lg[k] = expf(lg[k] - mx); s += lg[k]; }
    float inv = 1.0f / s;
    for (int k = 0; k < KTOP; ++k) att[h][k] = lg[k] * inv * mk[k];
  }
  __syncthreads();
  // V = hEV @ WVp, overwrite kv
  v8f d0 = {0,0,0,0,0,0,0,0}, d1 = {0,0,0,0,0,0,0,0};
  for (int k0 = 0; k0 < 256; k0 += 4) {
    int kk = k0 + 2*hf;
    v2f bf; bf.x = WVp[(size_t)kk*HID + n0 + lm];
            bf.y = WVp[(size_t)(kk+1)*HID + n0 + lm];
    v2f a0; a0.x = hEV[lm*256 + kk];      a0.y = hEV[lm*256 + kk + 1];
    v2f a1; a1.x = hEV[(16+lm)*256 + kk]; a1.y = hEV[(16+lm)*256 + kk + 1];
    d0 = wmma4(a0, bf, d0);
    d1 = wmma4(a1, bf, d1);
  }
  __syncthreads();
  #pragma unroll
  for (int r = 0; r < 8; ++r) {
    kv[(r + 8*hf)*HID + n0 + lm]      = d0[r];
    kv[(16 + r + 8*hf)*HID + n0 + lm] = d1[r];
  }
  __syncthreads();
  // dh = attn @ V
  if (tid < HID) {
    int h = tid >> 5;
    float acc = 0.0f;
    for (int k = 0; k < KTOP; ++k) acc += att[h][k] * kv[k*HID + tid];
    dh[tid] = acc;
  }
  __syncthreads();
  // @ WO + residual
  if (tid < HID) {
    float acc = 0.0f;
    for (int c = 0; c < HID; ++c) acc += dh[c] * WO[c*HID + tid];
    hmid[tid] = hv_in[(size_t)bi*HID + tid] + acc;
  }
  __syncthreads();
  block_ln_256(hmid, n1g, n1b, red, tid);
  // FFN
  for (int t = tid; t < 4*HID; t += 256) {
    float acc = Win_b[t];
    for (int c = 0; c < HID; ++c) acc += hmid[c] * Win[(size_t)c*4*HID + t];
    uarr[t] = fmaxf(acc, 0.0f);
  }
  __syncthreads();
  if (tid < HID) {
    float acc = Wout_b[tid];
    for (int j = 0; j < 4*HID; ++j) acc += uarr[j] * Wout[(size_t)j*HID + tid];
    dh[tid] = hmid[tid] + acc;
  }
  __syncthreads();
  block_ln_256(dh, n2g, n2b, red, tid);
  if (tid < HID) hv_out[(size_t)bi*HID + tid] = dh[tid];
}

__global__ void k_copy(float* __restrict__ dst, const float* __restrict__ src, size_t n) {
  size_t t = (size_t)blockIdx.x * blockDim.x + threadIdx.x;
  if (t < n) dst[t] = src[t];
}

// ---------------------------------------------------------------------------
// Host launch: resolve input pointers (insertion-order vs jax-alphabetical
// pytree flattening, detected from in_sizes), run pipeline on `stream`.
// ---------------------------------------------------------------------------
struct LayerW {
  const float *WQ, *WK, *WVp, *WO, *n1g, *n1b, *Win, *Winb, *Wout, *Woutb, *n2g, *n2b;
};

extern "C" void kernel_launch(void* const* d_in, const int* in_sizes, int n_in,
                              void* d_out, int out_size, void* d_ws, size_t ws_size,
                              hipStream_t stream) {
  auto FP = [&](int idx) { return (const float*)d_in[idx]; };
  auto IP = [&](int idx) { return (const int*)d_in[idx]; };

  const float *X, *mask, *node_W, *node_b, *node_g, *node_beta, *pos_W, *pos_b;
  const float *edge_W, *edge_b, *edge_g, *edge_beta, *Wv, *Wv_b, *We, *We_b;
  const int *srr;
  LayerW lw[3];

  bool insertion = (n_in > 0 && in_sizes[0] == NB * LSEQ * 4 * 3);  // X first?
  if (insertion) {
    X = FP(0); srr = IP(3); mask = FP(2);
    node_W = FP(4); node_b = FP(5); node_g = FP(6); node_beta = FP(7);
    pos_W = FP(8); pos_b = FP(9);
    edge_W = FP(10); edge_b = FP(11); edge_g = FP(12); edge_beta = FP(13);
    Wv = FP(14); Wv_b = FP(15); We = FP(16); We_b = FP(17);
    for (int l = 0; l < 3; ++l) {
      int o = 18 + 12*l;
      lw[l] = { FP(o+0), FP(o+1), FP(o+2), FP(o+3), FP(o+4), FP(o+5),
                FP(o+6), FP(o+7), FP(o+8), FP(o+9), FP(o+10), FP(o+11) };
    }
  } else {
    // jax tree_flatten (dict keys sorted): L, X, mask, We, We_b, Wv, Wv_b,
    // edge_W, edge_b, edge_beta, edge_g, layers[3]{WK,WO,WQ,WVp,Win,Win_b,
    // Wout,Wout_b,n1_b,n1_g,n2_b,n2_g}, node_W, node_b, node_beta, node_g,
    // pos_W, pos_b, single_res_rel
    X = FP(1); mask = FP(2);
    We = FP(3); We_b = FP(4); Wv = FP(5); Wv_b = FP(6);
    edge_W = FP(7); edge_b = FP(8); edge_beta = FP(9); edge_g = FP(10);
    for (int l = 0; l < 3; ++l) {
      int o = 11 + 12*l;
      lw[l].WK = FP(o+0); lw[l].WO = FP(o+1); lw[l].WQ = FP(o+2); lw[l].WVp = FP(o+3);
      lw[l].Win = FP(o+4); lw[l].Winb = FP(o+5); lw[l].Wout = FP(o+6); lw[l].Woutb = FP(o+7);
      lw[l].n1b = FP(o+8); lw[l].n1g = FP(o+9); lw[l].n2b = FP(o+10); lw[l].n2g = FP(o+11);
    }
    node_W = FP(47); node_b = FP(48); node_beta = FP(49); node_g = FP(50);
    pos_W = FP(51); pos_b = FP(52);
    srr = IP(53);
  }

  // workspace
  int*   E_idx = (int*)d_ws;
  float* hv0   = (float*)((char*)d_ws + 983040);
  float* h_E   = (float*)((char*)d_ws + 5177344);

  const int NBLK = NB * LSEQ;
  k_topk<<<NBLK, 128, 0, stream>>>(X, mask, E_idx);
  k_node<<<NBLK, 128, 0, stream>>>(X, node_W, node_b, node_g, node_beta, Wv, Wv_b, hv0);
  k_edge<<<NBLK, 256, 0, stream>>>(X, E_idx, srr, pos_W, pos_b,
                                   edge_W, edge_b, edge_g, edge_beta, We, We_b, h_E);

  float* out = (float*)d_out;
  const size_t N = (size_t)NB * LSEQ * HID;
  const float* hin = hv0;
  for (int l = 0; l < 3; ++l) {
    float* hout = out + N * (1 + l);   // hidden[l]
    k_layer<<<NBLK, 256, 0, stream>>>(hin, h_E, E_idx, mask,
        lw[l].WQ, lw[l].WK, lw[l].WVp, lw[l].WO, lw[l].n1g, lw[l].n1b,
        lw[l].Win, lw[l].Winb, lw[l].Wout, lw[l].Woutb, lw[l].n2g, lw[l].n2b,
        hout);
    hin = hout;
  }
  // out_feature = hidden[2]
  k_copy<<<(int)((N + 255) / 256), 256, 0, stream>>>(out, out + 3 * N, N);
}